// EntropyNetwork_62612033241362
// MI455X (gfx1250) — compile-verified
//
#include <hip/hip_runtime.h>

typedef float v2f __attribute__((ext_vector_type(2)));
typedef float v8f __attribute__((ext_vector_type(8)));

#define EPSV 0.01f
#define EXP_NEG_EPS 0.99004983374916811f   // exp(-0.01)
#define LOG2E 1.4426950408889634f
#define LN2   0.6931471805599453f

// NegativeLinear reparameterization: W>=0 ? -exp(-W-eps) : W - exp(-eps)
__device__ __forceinline__ float neg_reparam(float w) {
    return (w >= 0.0f) ? -__builtin_amdgcn_exp2f(-LOG2E * (w + EPSV))
                       : (w - EXP_NEG_EPS);
}

// -softplus(v) = -(max(v,0) + ln2*log2(1 + 2^(-|v|*log2e)))
// exp2 arg <= 0 -> e in (0,1] -> 1+e in (1,2]: no denorm/INF fixups needed,
// so the raw hardware trans builtins (v_exp_f32 / v_log_f32) are exact enough.
__device__ __forceinline__ float nsoftplus(float v) {
    float e = __builtin_amdgcn_exp2f(-LOG2E * __builtin_fabsf(v));
    float l = __builtin_amdgcn_logf(1.0f + e);       // log2(1+e)
    return __builtin_fmaf(-LN2, l, -__builtin_fmaxf(v, 0.0f));
}

// Phase-separated 8-wide activation: keeps the 8 independent trans chains
// interleaved so v_exp_f32/v_log_f32 pipeline with no hazard v_nops.
__device__ __forceinline__ v8f activate8(v8f c, float bias) {
    float t[8], e[8], l[8];
#pragma unroll
    for (int r = 0; r < 8; ++r) t[r] = c[r] + bias;
#pragma unroll
    for (int r = 0; r < 8; ++r)
        e[r] = __builtin_amdgcn_exp2f(-LOG2E * __builtin_fabsf(t[r]));
#pragma unroll
    for (int r = 0; r < 8; ++r)
        l[r] = __builtin_amdgcn_logf(1.0f + e[r]);
#pragma unroll
    for (int r = 0; r < 8; ++r)
        c[r] = __builtin_fmaf(-LN2, l[r], -__builtin_fmaxf(t[r], 0.0f));
    return c;
}

// Build one K=4 x N=16 fp32 B operand (2 VGPRs/lane) for y = x @ W^T:
// B[k][n] = W[n][k], zero outside (Kr, Nr).  B layout (32-bit, wave32):
// lanes 0-15 hold K = {k0, k0+1}, lanes 16-31 hold K = {k0+2, k0+3}, col = lane%16.
template <bool REPARAM>
__device__ __forceinline__ v2f load_B(const float* __restrict__ W,
                                      int Kr, int Nr, int k0, int lane) {
    int n  = lane & 15;
    int kb = k0 + ((lane >> 4) << 1);
    v2f b;
#pragma unroll
    for (int j = 0; j < 2; ++j) {
        int k   = kb + j;
        float w = 0.0f;
        if (k < Kr && n < Nr) {
            w = W[n * Kr + k];
            if (REPARAM) w = neg_reparam(w);
        }
        b[j] = w;
    }
    return b;
}

__device__ __forceinline__ v8f wmma4(v2f a, v2f b, v8f c) {
    // D(16x16,f32) = A(16x4,f32) * B(4x16,f32) + C
    return __builtin_amdgcn_wmma_f32_16x16x4_f32(
        /*neg_a=*/false, a, /*neg_b=*/false, b,
        /*c_mod=*/(short)0, c, /*reuse_a=*/false, /*reuse_b=*/false);
}

#define TSTRIDE 18                   // LDS row stride in floats (even -> 8B b64 loads)
#define TILE_FLOATS (16 * TSTRIDE)   // 288

__global__ __launch_bounds__(256) void entropy_net_kernel(
    const float* __restrict__ x0,
    const float* __restrict__ W_in,  const float* __restrict__ b_in,
    const float* __restrict__ W_out, const float* __restrict__ b_out,
    const float* __restrict__ Wl_out,const float* __restrict__ bl_out,
    const float* __restrict__ Wp1, const float* __restrict__ bp1,
    const float* __restrict__ Wl1, const float* __restrict__ bl1,
    const float* __restrict__ Wp2, const float* __restrict__ bp2,
    const float* __restrict__ Wl2, const float* __restrict__ bl2,
    const float* __restrict__ Wp3, const float* __restrict__ bp3,
    const float* __restrict__ Wl3, const float* __restrict__ bl3,
    const float* __restrict__ Wp4, const float* __restrict__ bp4,
    const float* __restrict__ Wl4, const float* __restrict__ bl4,
    float* __restrict__ out, int n)
{
    __shared__ __attribute__((aligned(16))) float lds[8 * 2 * TILE_FLOATS];

    const int lane = threadIdx.x & 31;
    const int wid  = threadIdx.x >> 5;
    const int ln   = lane & 15;     // tile row (point) / weight column index
    const int half = lane >> 4;     // lane half selects K sub-block

    float* tile0 = &lds[wid * 2 * TILE_FLOATS];
    float* tile1 = tile0 + TILE_FLOATS;

    const float* Wp[4] = {Wp1, Wp2, Wp3, Wp4};
    const float* Wl[4] = {Wl1, Wl2, Wl3, Wl4};
    const float* bp[4] = {bp1, bp2, bp3, bp4};
    const float* bl[4] = {bl1, bl2, bl3, bl4};

    // ---- wave-resident, pre-swizzled B operands (weights) ----
    v2f Bin = load_B<false>(W_in, 3, 7, 0, lane);       // [N,3]x[3,7]
    v2f Bp_lo[4], Bp_hi[4], Bl_[4];
    float bias_blk[4];
#pragma unroll
    for (int i = 0; i < 4; ++i) {
        Bp_lo[i] = load_B<true >(Wp[i], 7, 7, 0, lane); // reparam 7x7, K 0..3
        Bp_hi[i] = load_B<true >(Wp[i], 7, 7, 4, lane); // K 4..7 (row 7 = 0)
        Bl_[i]   = load_B<false>(Wl[i], 3, 7, 0, lane); // skip term [N,3]x[3,7]
        bias_blk[i] = (ln < 7) ? (bp[i][ln] + bl[i][ln]) : 0.0f;
    }
    float bias_in = (ln < 7) ? b_in[ln] : 0.0f;

    // final layer weights (broadcast, per-lane registers)
    float wout[7], wlout[3];
#pragma unroll
    for (int k = 0; k < 7; ++k) wout[k] = neg_reparam(W_out[k]);
#pragma unroll
    for (int k = 0; k < 3; ++k) wlout[k] = Wl_out[k];
    float bias_out = b_out[0] + bl_out[0];

    const v8f z = {0, 0, 0, 0, 0, 0, 0, 0};

    const int nwaves = (int)((gridDim.x * blockDim.x) >> 5);
    const int gwave  = (int)((blockIdx.x * blockDim.x + threadIdx.x) >> 5);
    const int pairs  = n >> 5;   // each wave iteration = two 16-point tiles

    for (int pair = gwave; pair < pairs; pair += nwaves) {
        const int base = pair << 5;
        v2f ax0_t[2];

#pragma unroll
        for (int t = 0; t < 2; ++t) {
            float* tile = t ? tile1 : tile0;
            const int pt = base + (t << 4) + ln;
            const float* xp = x0 + pt * 3;

            // x0 as A operand: half 0 -> (x,y) = K0,K1 ; half 1 -> (z,0) = K2,K3
            v2f a0;
            if (half == 0) { a0[0] = xp[0]; a0[1] = xp[1]; }
            else           { a0[0] = xp[2]; a0[1] = 0.0f;  }
            ax0_t[t] = a0;

            // ---- input layer: x = -sp(x0 @ W_in^T + b_in) ----
            v8f c = wmma4(a0, Bin, z);
            c = activate8(c, bias_in);
#pragma unroll
            for (int r = 0; r < 8; ++r)
                tile[(r + half * 8) * TSTRIDE + ln] = c[r];   // C layout -> (M,N)
            asm volatile("" ::: "memory");  // keep DS stores before next DS loads

            // ---- 4 residual-style blocks ----
#pragma unroll
            for (int i = 0; i < 4; ++i) {
                const int k0 = half * 2;
                // A operands from LDS (re-swizzle C -> A): K 0..3 and K 4..7
                v2f alo = *(const v2f*)&tile[ln * TSTRIDE + k0];
                v2f ahi = *(const v2f*)&tile[ln * TSTRIDE + 4 + k0];
                asm volatile("" ::: "memory");

                v8f acc = wmma4(alo, Bp_lo[i], z);
                acc = wmma4(ahi,      Bp_hi[i], acc);
                acc = wmma4(ax0_t[t], Bl_[i],   acc);
                acc = activate8(acc, bias_blk[i]);
#pragma unroll
                for (int r = 0; r < 8; ++r)
                    tile[(r + half * 8) * TSTRIDE + ln] = acc[r];
                asm volatile("" ::: "memory");
            }
        }

        // ---- output layer: one point per lane across both tiles ----
        {
            const float* tile = (lane < 16) ? tile0 : tile1;
            const int pt = base + lane;
            float s = bias_out;
#pragma unroll
            for (int k = 0; k < 7; ++k) s += tile[ln * TSTRIDE + k] * wout[k];
            const float* xp = x0 + pt * 3;
#pragma unroll
            for (int k = 0; k < 3; ++k) s += xp[k] * wlout[k];
            out[pt] = nsoftplus(s);
        }
    }
}

extern "C" void kernel_launch(void* const* d_in, const int* in_sizes, int n_in,
                              void* d_out, int out_size, void* d_ws, size_t ws_size,
                              hipStream_t stream) {
    const float* x0     = (const float*)d_in[0];
    const float* W_in   = (const float*)d_in[1];
    const float* b_in   = (const float*)d_in[2];
    const float* W_out  = (const float*)d_in[3];
    const float* b_out  = (const float*)d_in[4];
    const float* Wl_out = (const float*)d_in[5];
    const float* bl_out = (const float*)d_in[6];
    const float* Wp1 = (const float*)d_in[7];
    const float* bp1 = (const float*)d_in[8];
    const float* Wl1 = (const float*)d_in[9];
    const float* bl1 = (const float*)d_in[10];
    const float* Wp2 = (const float*)d_in[11];
    const float* bp2 = (const float*)d_in[12];
    const float* Wl2 = (const float*)d_in[13];
    const float* bl2 = (const float*)d_in[14];
    const float* Wp3 = (const float*)d_in[15];
    const float* bp3 = (const float*)d_in[16];
    const float* Wl3 = (const float*)d_in[17];
    const float* bl3 = (const float*)d_in[18];
    const float* Wp4 = (const float*)d_in[19];
    const float* bp4 = (const float*)d_in[20];
    const float* Wl4 = (const float*)d_in[21];
    const float* bl4 = (const float*)d_in[22];
    float* out = (float*)d_out;

    const int n = in_sizes[0] / 3;   // number of points (N)

    dim3 grid(1024), block(256);     // 8192 waves; 131072 pair-tiles -> 16 iters/wave
    entropy_net_kernel<<<grid, block, 0, stream>>>(
        x0, W_in, b_in, W_out, b_out, Wl_out, bl_out,
        Wp1, bp1, Wl1, bl1, Wp2, bp2, Wl2, bl2,
        Wp3, bp3, Wl3, bl3, Wp4, bp4, Wl4, bl4,
        out, n);
}